// QuantizedWeight_77919296684642
// MI455X (gfx1250) — compile-verified
//
#include <hip/hip_runtime.h>

typedef __attribute__((ext_vector_type(2))) float v2f;
typedef __attribute__((ext_vector_type(8))) float v8f;

#define ROWS 4096
#define COLS 11008
#define RANK 32
#define CPR  (COLS / 8)            // codes per weight row = 1376
#define CT   4                     // column tiles (16 cols each) per wave
#define WPB  4                     // waves per block
#define NCTG ((COLS / 16) / CT)    // 688/4 = 172 column-tile groups
#define NTASK ((ROWS / 16) * NCTG) // 256 * 172 = 44032 wave tasks

// out[r*COLS+c] = codebooks[cb][codes_flat[g]][e] * scales[i>>12] + (L@R)[r][c]
//   i = r*COLS+c, g = i>>3 = r*CPR + c/8, e = c&7, cb = r>>11
__global__ __launch_bounds__(WPB * 32)
void aqlm_dequant_lowrank(const float* __restrict__ codebooks,
                          const int*   __restrict__ codes,
                          const float* __restrict__ scales,
                          const float* __restrict__ Lm,   // (4096, 32)
                          const float* __restrict__ Rm,   // (32, 11008)
                          float*       __restrict__ out)  // (4096, 11008)
{
    // Stage both codebooks (2*256*8 f32 = 16 KB) into LDS.
    __shared__ float cbLds[2 * 256 * 8];
    for (int t = threadIdx.x; t < 2 * 256 * 8; t += WPB * 32)
        cbLds[t] = codebooks[t];
    __syncthreads();

    const int lane = threadIdx.x & 31;
    const int wave = threadIdx.x >> 5;
    const int task = blockIdx.x * WPB + wave;       // grid sized exactly
    const int rowTile = task / NCTG;
    const int ctGroup = task - rowTile * NCTG;
    const int r0 = rowTile * 16;
    const int c0base = ctGroup * (16 * CT);

    const int nl = lane & 15;   // M (for A) / N (for B,C,D)
    const int hl = lane >> 4;   // lane-half selects K pair / M+8
    const int kb = hl * 2;

    // A fragments for all 8 K-steps (ISA 16x4 f32 A layout:
    // lanes 0-15 M=0..15 hold K={0,1}, lanes 16-31 hold K={2,3}).
    v2f a[8];
    const float* Lrow = Lm + (r0 + nl) * RANK + kb;
    #pragma unroll
    for (int s = 0; s < 8; ++s) {
        a[s].x = Lrow[4 * s];
        a[s].y = Lrow[4 * s + 1];
    }

    // Codebook select is uniform per 16-row tile.
    const float* cbBase = cbLds + ((r0 >> 11) << 11);   // + cb*2048

    for (int ct = 0; ct < CT; ++ct) {
        const int c0 = c0base + ct * 16;

        // Low-rank correction: D = L[r0:+16, :] @ R[:, c0:+16] via 8x K=4 WMMA.
        v8f acc = {};
        const float* Rp = Rm + kb * COLS + c0 + nl;     // B: 4x16 mirrored layout
        #pragma unroll
        for (int s = 0; s < 8; ++s) {
            v2f b;
            b.x = Rp[(4 * s)     * COLS];
            b.y = Rp[(4 * s + 1) * COLS];
            acc = __builtin_amdgcn_wmma_f32_16x16x4_f32(
                false, a[s], false, b, (short)0, acc, false, false);
        }

        // 32 distinct code entries per tile: lane l holds
        // (row = r0 + (l&15), code-col = c0/8 + (l>>4)).
        const int myCode = codes[(r0 + nl) * CPR + (c0 >> 3) + hl];

        // C/D layout: VGPR v <-> row (v + 8*hl), col nl.
        #pragma unroll
        for (int v = 0; v < 8; ++v) {
            const int rr = r0 + v + 8 * hl;
            const unsigned i = (unsigned)rr * COLS + (unsigned)(c0 + nl);
            const int srcLane = (v + 8 * hl) | ((nl >> 3) << 4);
            const int code = __builtin_amdgcn_ds_bpermute(srcLane << 2, myCode);
            const float w = cbBase[code * 8 + (nl & 7)];
            out[i] = fmaf(w, scales[i >> 12], acc[v]);
        }
    }
}

extern "C" void kernel_launch(void* const* d_in, const int* in_sizes, int n_in,
                              void* d_out, int out_size, void* d_ws, size_t ws_size,
                              hipStream_t stream) {
    (void)in_sizes; (void)n_in; (void)out_size; (void)d_ws; (void)ws_size;
    const float* codebooks = (const float*)d_in[0];  // (2,256,8) f32
    const int*   codes     = (const int*)d_in[1];    // (2, 2818048) i32
    const float* scales    = (const float*)d_in[2];  // (11008,1) f32
    const float* Lm        = (const float*)d_in[3];  // (4096,32) f32
    const float* Rm        = (const float*)d_in[4];  // (32,11008) f32
    float* out = (float*)d_out;                      // (4096,11008) f32

    dim3 grid(NTASK / WPB);      // 11008 blocks
    dim3 block(WPB * 32);        // 128 threads = 4 waves
    aqlm_dequant_lowrank<<<grid, block, 0, stream>>>(codebooks, codes, scales,
                                                     Lm, Rm, out);
}